// resconv_13237089206323
// MI455X (gfx1250) — compile-verified
//
#include <hip/hip_runtime.h>

// Problem constants (match reference)
#define BB 2
#define CC 64
#define DD 3
#define NN 4096
#define KK 16
#define EE 128   // 2*C  edge channels
#define HH 256   // 2*E  hidden channels

typedef __attribute__((ext_vector_type(2))) float v2f;
typedef __attribute__((ext_vector_type(8))) float v8f;

constexpr int NT   = 4;          // n-values per workgroup
constexpr int COLS = NT * KK;    // 64 columns: col = nl*16 + k
constexpr int LDSC = COLS + 4;   // padded LDS row stride (floats)

// ---------------------------------------------------------------------------
// Kernel 1: mean over D  ->  mean_pc (B,C,N)
// ---------------------------------------------------------------------------
__global__ void meanpc_kernel(const float* __restrict__ x, float* __restrict__ mean_pc) {
    int i = blockIdx.x * blockDim.x + threadIdx.x;        // over B*C*N
    if (i >= BB * CC * NN) return;
    int n  = i % NN;
    int bc = i / NN;
    const float* xp = x + (size_t)bc * (DD * NN) + n;
    mean_pc[i] = (xp[0] + xp[NN] + xp[2 * NN]) * (1.0f / 3.0f);
}

// ---------------------------------------------------------------------------
// Kernel 2: squared norms of mean_pc columns -> sq (B,N)
// ---------------------------------------------------------------------------
__global__ void sqnorm_kernel(const float* __restrict__ mean_pc, float* __restrict__ sq) {
    int i = blockIdx.x * blockDim.x + threadIdx.x;        // over B*N
    if (i >= BB * NN) return;
    int n = i % NN;
    int b = i / NN;
    float s = 0.0f;
#pragma unroll
    for (int c = 0; c < CC; ++c) {
        float v = mean_pc[((size_t)b * CC + c) * NN + n];
        s = __builtin_fmaf(v, v, s);
    }
    sq[i] = s;
}

// ---------------------------------------------------------------------------
// Kernel 3: brute-force KNN (top-16 nearest, self excluded, nearest-first)
// one query per lane; reference point loads are wave-uniform (broadcast)
// ---------------------------------------------------------------------------
__global__ void __launch_bounds__(256) knn_kernel(const float* __restrict__ mean_pc,
                                                  const float* __restrict__ sq,
                                                  int* __restrict__ idx) {
    int t = threadIdx.x;
    int b = blockIdx.x / (NN / 256);
    int n = (blockIdx.x % (NN / 256)) * 256 + t;

    float q[CC];
#pragma unroll
    for (int c = 0; c < CC; ++c) q[c] = mean_pc[((size_t)b * CC + c) * NN + n];

    float bd[KK];
    int   bi[KK];
#pragma unroll
    for (int j = 0; j < KK; ++j) { bd[j] = 3.0e38f; bi[j] = 0; }

    for (int r = 0; r < NN; ++r) {
        float dot = 0.0f;
#pragma unroll
        for (int c = 0; c < CC; ++c)
            dot = __builtin_fmaf(q[c], mean_pc[((size_t)b * CC + c) * NN + r], dot);
        // d2 = |q|^2 + |r|^2 - 2*dot ; |q|^2 constant per thread -> drop it
        float d2 = sq[(size_t)b * NN + r] - 2.0f * dot;
        if (r != n && d2 < bd[KK - 1]) {
            bd[KK - 1] = d2;
            bi[KK - 1] = r;
            // single bubble pass restores sorted order (fully unrolled -> registers)
#pragma unroll
            for (int j = KK - 1; j > 0; --j) {
                if (bd[j] < bd[j - 1]) {
                    float td = bd[j]; bd[j] = bd[j - 1]; bd[j - 1] = td;
                    int   ti = bi[j]; bi[j] = bi[j - 1]; bi[j - 1] = ti;
                }
            }
        }
    }
#pragma unroll
    for (int j = 0; j < KK; ++j) idx[((size_t)b * KK + j) * NN + n] = bi[j];
}

// ---------------------------------------------------------------------------
// WMMA f32 16x16x4 tile engine.
//   A fragment (16x4, W row-major, fp32): vgpr j, lane l -> W[rt*16 + (l&15)][k0 + j + 2*(l>>4)]
//   B fragment (4x16, from LDS):          vgpr j, lane l -> In[k0 + j + 2*(l>>4)][ct*16 + (l&15)]
//   D (16x16): vgpr j, lane l -> row rt*16 + j + 8*(l>>4), col ct*16 + (l&15)
// ---------------------------------------------------------------------------
template <int NCT, bool RELU>
__device__ __forceinline__ void gemm_tile(const float* __restrict__ W,
                                          const float* __restrict__ bias,
                                          const float* lin, float* lout,
                                          int Ki, int rt, int ctBase,
                                          int laneRow, int laneHi) {
    const v8f vzero = {0.f, 0.f, 0.f, 0.f, 0.f, 0.f, 0.f, 0.f};
    v8f acc[NCT];
#pragma unroll
    for (int c = 0; c < NCT; ++c) acc[c] = vzero;

    const float* wrow = W + (size_t)(rt * 16 + laneRow) * Ki;
    for (int k0 = 0; k0 < Ki; k0 += 4) {
        v2f a;
        a.x = wrow[k0 + 2 * laneHi + 0];
        a.y = wrow[k0 + 2 * laneHi + 1];
#pragma unroll
        for (int c = 0; c < NCT; ++c) {
            int col = (ctBase + c) * 16 + laneRow;
            v2f bf;
            bf.x = lin[(k0 + 2 * laneHi + 0) * LDSC + col];
            bf.y = lin[(k0 + 2 * laneHi + 1) * LDSC + col];
            acc[c] = __builtin_amdgcn_wmma_f32_16x16x4_f32(
                false, a, false, bf, (short)0, acc[c], false, false);
        }
    }
    // fused bias (+ReLU) epilogue, write D tile to LDS
#pragma unroll
    for (int c = 0; c < NCT; ++c) {
#pragma unroll
        for (int j = 0; j < 8; ++j) {
            int row = rt * 16 + j + 8 * laneHi;
            float v = acc[c][j] + bias[row];
            if (RELU) v = fmaxf(v, 0.0f);
            lout[row * LDSC + (ctBase + c) * 16 + laneRow] = v;
        }
    }
}

// ---------------------------------------------------------------------------
// Kernel 4: fused edge-build + 3-layer MLP + max_k + residual
// One WG = (b, d, 4 n-values) x all 16 k  ->  64 columns, col = nl*16 + k
// ---------------------------------------------------------------------------
__global__ void __launch_bounds__(256) mlp_kernel(const float* __restrict__ x,
                                                  const float* __restrict__ mean_pc,
                                                  const int* __restrict__ idx,
                                                  const float* __restrict__ W1, const float* __restrict__ b1,
                                                  const float* __restrict__ W2, const float* __restrict__ b2,
                                                  const float* __restrict__ W3, const float* __restrict__ b3,
                                                  float* __restrict__ out) {
    // Two ping-pong regions: X holds edge(128 rows) then h2(256); Y holds h1(256) then r(64)
    __shared__ float ldsX[HH * LDSC];
    __shared__ float ldsY[HH * LDSC];

    int t       = threadIdx.x;
    int lane    = t & 31;
    int w       = t >> 5;          // wave id 0..7
    int laneRow = lane & 15;
    int laneHi  = lane >> 4;

    int blk = blockIdx.x;          // over B*D*(N/NT)
    int nb  = blk % (NN / NT);
    int bd  = blk / (NN / NT);
    int d   = bd % DD;
    int b   = bd / DD;
    int n0  = nb * NT;

    // ---- stage edge matrix (128 x 64) into ldsX; fuses the neighbor gather ----
    for (int e = t; e < EE * COLS; e += 256) {
        int row = e >> 6;          // / COLS
        int col = e & (COLS - 1);
        int nl  = col >> 4;
        int k   = col & 15;
        int n   = n0 + nl;
        float v;
        if (row < CC) {
            v = x[(((size_t)b * CC + row) * DD + d) * NN + n];
        } else {
            int c = row - CC;
            int j = idx[((size_t)b * KK + k) * NN + n];
            v = mean_pc[((size_t)b * CC + c) * NN + j]
              - x[(((size_t)b * CC + c) * DD + d) * NN + n];
        }
        ldsX[row * LDSC + col] = v;
    }
    __syncthreads();

    // ---- L1: h1 = relu(W1(256x128) @ edge + b1) -> ldsY. 16 row-tiles / 8 waves ----
    gemm_tile<4, true>(W1, b1, ldsX, ldsY, EE, 2 * w + 0, 0, laneRow, laneHi);
    gemm_tile<4, true>(W1, b1, ldsX, ldsY, EE, 2 * w + 1, 0, laneRow, laneHi);
    __syncthreads();

    // ---- L2: h2 = relu(W2(256x256) @ h1 + b2) -> ldsX (edge dead) ----
    gemm_tile<4, true>(W2, b2, ldsY, ldsX, HH, 2 * w + 0, 0, laneRow, laneHi);
    gemm_tile<4, true>(W2, b2, ldsY, ldsX, HH, 2 * w + 1, 0, laneRow, laneHi);
    __syncthreads();

    // ---- L3: r = W3(64x256) @ h2 + b3 -> ldsY (h1 dead). 4 rt x 4 ct / 8 waves ----
    gemm_tile<2, false>(W3, b3, ldsX, ldsY, HH, w >> 1, (w & 1) * 2, laneRow, laneHi);
    __syncthreads();

    // ---- epilogue: out[b,c,d,n] = max_k r[c, nl*16+k] + x[b,c,d,n] ----
    {
        int c  = t & (CC - 1);
        int nl = t >> 6;           // 0..3
        const float* rrow = &ldsY[c * LDSC + nl * 16];
        float m = rrow[0];
#pragma unroll
        for (int k = 1; k < KK; ++k) m = fmaxf(m, rrow[k]);
        size_t o = (((size_t)b * CC + c) * DD + d) * NN + (n0 + nl);
        out[o] = m + x[o];
    }
}

// ---------------------------------------------------------------------------
// Host launcher
// ---------------------------------------------------------------------------
extern "C" void kernel_launch(void* const* d_in, const int* in_sizes, int n_in,
                              void* d_out, int out_size, void* d_ws, size_t ws_size,
                              hipStream_t stream) {
    const float* x  = (const float*)d_in[0];
    const float* W1 = (const float*)d_in[1];
    const float* b1 = (const float*)d_in[2];
    const float* W2 = (const float*)d_in[3];
    const float* b2 = (const float*)d_in[4];
    const float* W3 = (const float*)d_in[5];
    const float* b3 = (const float*)d_in[6];
    float* out = (float*)d_out;

    // workspace carve-out
    float* mean_pc = (float*)d_ws;                    // B*C*N floats
    float* sq      = mean_pc + (size_t)BB * CC * NN;  // B*N floats
    int*   idxp    = (int*)(sq + (size_t)BB * NN);    // B*K*N ints

    meanpc_kernel<<<(BB * CC * NN + 255) / 256, 256, 0, stream>>>(x, mean_pc);
    sqnorm_kernel<<<(BB * NN + 255) / 256, 256, 0, stream>>>(mean_pc, sq);
    knn_kernel<<<BB * (NN / 256), 256, 0, stream>>>(mean_pc, sq, idxp);
    mlp_kernel<<<BB * DD * (NN / NT), 256, 0, stream>>>(x, mean_pc, idxp,
                                                        W1, b1, W2, b2, W3, b3, out);
}